// DynamicSparseConv_86492051407246
// MI455X (gfx1250) — compile-verified
//
#include <hip/hip_runtime.h>
#include <hip/hip_bf16.h>
#include <math.h>

// Problem constants (from reference): B=32, C=256, H=W=64, MID=32, 9*C=2304
#define Bn   32
#define Cn   256
#define HWn  4096        // 64*64
#define MIDn 32
#define NOUT 2304        // 9*C

typedef __attribute__((ext_vector_type(2))) float v2f;
typedef __attribute__((ext_vector_type(8))) float v8f;

// ---------------------------------------------------------------------------
// Kernel 1: per-(b,c) channel mean over the 64x64 plane.
// ---------------------------------------------------------------------------
__global__ __launch_bounds__(256) void mean_kernel(const float* __restrict__ x,
                                                   float* __restrict__ y) {
  const int bc  = blockIdx.x;          // 0 .. B*C-1
  const int tid = threadIdx.x;
  const float4* xp = (const float4*)(x + (size_t)bc * HWn);   // 1024 float4
  float s = 0.0f;
  #pragma unroll
  for (int q = tid; q < 1024; q += 256) {
    float4 v = xp[q];
    s += v.x + v.y + v.z + v.w;
  }
  __shared__ float red[256];
  red[tid] = s;
  __syncthreads();
  for (int off = 128; off > 0; off >>= 1) {
    if (tid < off) red[tid] += red[tid + off];
    __syncthreads();
  }
  if (tid == 0) y[bc] = red[0] * (1.0f / (float)HWn);
}

// ---------------------------------------------------------------------------
// Kernel 2: h = gelu(y @ w1^T)  — [32,256]x[256,32] -> [32,32]
// One workgroup, 4 waves; each wave owns a 16x16 tile; K=256 in steps of 4
// via V_WMMA_F32_16X16X4_F32.
// A layout (16x4 f32): lane(m=L&15, half=L>>4) -> {A[m][k0+2h], A[m][k0+2h+1]}
// B layout symmetric (N across lanes), C/D per ISA 8-VGPR 16x16 layout.
// ---------------------------------------------------------------------------
__global__ __launch_bounds__(128) void gemm1_gelu_kernel(const float* __restrict__ y,
                                                         const float* __restrict__ w1,
                                                         float* __restrict__ h) {
  const int wid  = threadIdx.x >> 5;        // 0..3
  const int lane = threadIdx.x & 31;
  const int mt   = (wid >> 1) * 16;         // 0 or 16 (batch tile)
  const int nt   = (wid & 1) * 16;          // 0 or 16 (MID tile)
  const int half = lane >> 4;               // 0/1
  const int r    = lane & 15;

  const float* arow = y  + (size_t)(mt + r) * Cn;   // Y[m][k]
  const float* brow = w1 + (size_t)(nt + r) * Cn;   // B[k][n] = w1[n][k]

  v8f acc = {};
  for (int k0 = 0; k0 < Cn; k0 += 4) {
    const int ka = k0 + half * 2;
    v2f a; a.x = arow[ka]; a.y = arow[ka + 1];
    v2f b; b.x = brow[ka]; b.y = brow[ka + 1];
    acc = __builtin_amdgcn_wmma_f32_16x16x4_f32(false, a, false, b,
                                                (short)0, acc, false, false);
  }
  #pragma unroll
  for (int vr = 0; vr < 8; ++vr) {
    const int row = mt + vr + half * 8;     // C/D: VGPR vr holds M=vr (+8 for hi lanes)
    const int col = nt + r;
    const float v = acc[vr];
    h[row * MIDn + col] = 0.5f * v * (1.0f + erff(v * 0.70710678118654752f));
  }
}

// ---------------------------------------------------------------------------
// Kernel 3: wraw = h @ w2^T — [32,32]x[32,2304] -> [32,2304]
// Grid = 2 (M tiles) x 144 (N tiles) = 288 blocks of one wave each.
// ---------------------------------------------------------------------------
__global__ __launch_bounds__(32) void gemm2_kernel(const float* __restrict__ h,
                                                   const float* __restrict__ w2,
                                                   float* __restrict__ wraw) {
  const int mt    = (blockIdx.x & 1) * 16;
  const int nbase = (blockIdx.x >> 1) * 16;      // 0..143 tiles of the 2304 dim
  const int lane  = threadIdx.x;
  const int half  = lane >> 4;
  const int r     = lane & 15;

  const float* arow = h  + (size_t)(mt + r) * MIDn;     // H[m][k]
  const float* brow = w2 + (size_t)(nbase + r) * MIDn;  // B[k][n] = w2[n][k]

  v8f acc = {};
  #pragma unroll
  for (int k0 = 0; k0 < MIDn; k0 += 4) {
    const int ka = k0 + half * 2;
    v2f a; a.x = arow[ka]; a.y = arow[ka + 1];
    v2f b; b.x = brow[ka]; b.y = brow[ka + 1];
    acc = __builtin_amdgcn_wmma_f32_16x16x4_f32(false, a, false, b,
                                                (short)0, acc, false, false);
  }
  #pragma unroll
  for (int vr = 0; vr < 8; ++vr) {
    const int row = mt + vr + half * 8;
    wraw[(size_t)row * NOUT + nbase + r] = acc[vr];
  }
}

// ---------------------------------------------------------------------------
// Kernel 4: per-(b,c) softmax over the 9 kernel taps.
// ---------------------------------------------------------------------------
__global__ __launch_bounds__(256) void softmax9_kernel(const float* __restrict__ wraw,
                                                       float* __restrict__ wts) {
  const int idx = blockIdx.x * blockDim.x + threadIdx.x;   // 0..8191
  const int b = idx >> 8;
  const int c = idx & 255;
  const float* p = wraw + (size_t)b * NOUT + c * 9;
  float v[9];
  float m = -INFINITY;
  #pragma unroll
  for (int k = 0; k < 9; ++k) { v[k] = p[k]; m = fmaxf(m, v[k]); }
  float s = 0.0f;
  #pragma unroll
  for (int k = 0; k < 9; ++k) { v[k] = expf(v[k] - m); s += v[k]; }
  const float inv = 1.0f / s;
  float* q = wts + (size_t)idx * 9;
  #pragma unroll
  for (int k = 0; k < 9; ++k) q[k] = v[k] * inv;
}

// ---------------------------------------------------------------------------
// Kernel 5: dynamic depthwise 3x3 conv (pad=1) + residual. Memory-bound pass:
// one block per (b,c) plane; 66x68-padded LDS tile with zero halo; float4
// global loads/stores for full 128-bit VMEM transactions.
// ---------------------------------------------------------------------------
__global__ __launch_bounds__(256) void dwconv_kernel(const float* __restrict__ x,
                                                     const float* __restrict__ wts,
                                                     float* __restrict__ out) {
  __shared__ float s[66][68];      // row stride 68 -> no 64-bank conflicts
  __shared__ float wk[9];
  const int bc  = blockIdx.x;
  const int tid = threadIdx.x;
  const float*  xp = x   + (size_t)bc * HWn;
  float*        op = out + (size_t)bc * HWn;

  // zero whole tile (covers the pad=1 halo), fetch this plane's 9 weights
  for (int i = tid; i < 66 * 68; i += 256) (&s[0][0])[i] = 0.0f;
  if (tid < 9) wk[tid] = wts[(size_t)bc * 9 + tid];
  __syncthreads();

  // interior fill: 1024 float4 per plane
  #pragma unroll
  for (int q = tid; q < 1024; q += 256) {
    const float4 v = ((const float4*)xp)[q];
    const int i = q >> 4;            // row 0..63
    const int j = (q & 15) << 2;     // col 0..60 step 4
    s[i + 1][j + 1] = v.x;
    s[i + 1][j + 2] = v.y;
    s[i + 1][j + 3] = v.z;
    s[i + 1][j + 4] = v.w;
  }
  __syncthreads();

  const float w0 = wk[0], w1 = wk[1], w2 = wk[2];
  const float w3 = wk[3], w4 = wk[4], w5 = wk[5];
  const float w6 = wk[6], w7 = wk[7], w8 = wk[8];

  #pragma unroll
  for (int q = tid; q < 1024; q += 256) {
    const int i = q >> 4;
    const int j = (q & 15) << 2;
    float o[4];
    #pragma unroll
    for (int t = 0; t < 4; ++t) {
      const int jc = j + t;
      const float cen = s[i + 1][jc + 1];
      float acc = w0 * s[i    ][jc    ] + w1 * s[i    ][jc + 1] + w2 * s[i    ][jc + 2]
                + w3 * s[i + 1][jc    ] + w4 * cen              + w5 * s[i + 1][jc + 2]
                + w6 * s[i + 2][jc    ] + w7 * s[i + 2][jc + 1] + w8 * s[i + 2][jc + 2];
      o[t] = acc + cen;   // residual
    }
    ((float4*)op)[q] = make_float4(o[0], o[1], o[2], o[3]);
  }
}

// ---------------------------------------------------------------------------
// Host launcher
// ---------------------------------------------------------------------------
extern "C" void kernel_launch(void* const* d_in, const int* in_sizes, int n_in,
                              void* d_out, int out_size, void* d_ws, size_t ws_size,
                              hipStream_t stream) {
  const float* x  = (const float*)d_in[0];   // [32,256,64,64]
  const float* w1 = (const float*)d_in[1];   // [32,256]
  const float* w2 = (const float*)d_in[2];   // [2304,32]
  float* out = (float*)d_out;

  float* ws   = (float*)d_ws;
  float* y    = ws;                         // 8192 floats
  float* h    = y + Bn * Cn;                // 1024 floats
  float* wraw = h + Bn * MIDn;              // 73728 floats
  float* wts  = wraw + Bn * NOUT;           // 73728 floats

  mean_kernel     <<<Bn * Cn, 256, 0, stream>>>(x, y);
  gemm1_gelu_kernel<<<1, 128, 0, stream>>>(y, w1, h);
  gemm2_kernel    <<<2 * (NOUT / 16), 32, 0, stream>>>(h, w2, wraw);
  softmax9_kernel <<<(Bn * Cn) / 256, 256, 0, stream>>>(wraw, wts);
  dwconv_kernel   <<<Bn * Cn, 256, 0, stream>>>(x, wts, out);
}